// CachedSddmmLinear_28192165331682
// MI455X (gfx1250) — compile-verified
//
#include <hip/hip_runtime.h>
#include <hip/hip_bf16.h>

typedef __attribute__((ext_vector_type(2))) float v2f;
typedef __attribute__((ext_vector_type(8))) float v8f;

#define IN_FEATURES 4096
#define OUT_FEATURES 14336
#define TOPK 1228

// ---------------------------------------------------------------------------
// Kernel 1: exact radix-select of the TOPK-th largest |x| (bit-pattern order
// == float order for non-negative floats), then write masked x to workspace.
// Single workgroup of 1024 threads; 4 passes of 8 bits, MSB first.
// ---------------------------------------------------------------------------
__global__ __launch_bounds__(1024)
void topk_mask_kernel(const float* __restrict__ x, float* __restrict__ xm) {
  __shared__ unsigned hist[256];
  __shared__ unsigned s_pref;
  __shared__ int s_rem;
  __shared__ unsigned s_eq;
  const int tid = threadIdx.x;
  if (tid == 0) { s_pref = 0u; s_rem = TOPK; s_eq = 0u; }
  __syncthreads();

  for (int pass = 0; pass < 4; ++pass) {
    const int shift = 24 - 8 * pass;
    if (tid < 256) hist[tid] = 0u;
    __syncthreads();
    const unsigned pref = s_pref;
    for (int i = tid; i < IN_FEATURES; i += 1024) {
      unsigned key = __float_as_uint(x[i]) & 0x7fffffffu;
      bool match = (pass == 0) || ((key >> (shift + 8)) == pref);
      if (match) atomicAdd(&hist[(key >> shift) & 0xffu], 1u);
    }
    __syncthreads();
    if (tid == 0) {
      int rem = s_rem;
      int acc = 0;
      int b = 255;
      for (; b >= 0; --b) {
        int h = (int)hist[b];
        if (acc + h >= rem) { s_rem = rem - acc; break; }
        acc += h;
      }
      if (b < 0) b = 0;
      s_pref = (s_pref << 8) | (unsigned)b;
    }
    __syncthreads();
  }

  const unsigned T = s_pref;  // key of the k-th largest element
  const int rem = s_rem;      // number of ==T elements still to admit
  for (int i = tid; i < IN_FEATURES; i += 1024) {
    unsigned key = __float_as_uint(x[i]) & 0x7fffffffu;
    bool sel = key > T;
    if (key == T) sel = (atomicAdd(&s_eq, 1u) < (unsigned)rem);
    xm[i] = sel ? x[i] : 0.0f;
  }
}

// ---------------------------------------------------------------------------
// Kernel 2: dense masked GEMV via V_WMMA_F32_16X16X4_F32.
// Block = 256 threads = 8 waves; each wave owns a 16-row tile of W.
// Grid = 14336 / 128 = 112 blocks. x (masked) staged in LDS, broadcast into
// all 16 B-columns so each wmma computes 16 row-dot-products (all N equal).
// A layout (16x4 f32): lane m in 0..15 -> row m; VGPR0/1 = K {0,1} for lanes
// 0-15 and K {2,3} for lanes 16-31  =>  per-lane b64 load at k + (lane>>4)*2.
// B layout mirrors this in K; value depends only on K (broadcast), so a b64
// ds_load at the same K offset is correct regardless of the N striping.
// ---------------------------------------------------------------------------
__global__ __launch_bounds__(256)
void gemv_wmma_kernel(const float* __restrict__ W,
                      const float* __restrict__ bias,
                      const float* __restrict__ xm,
                      float* __restrict__ out) {
  __shared__ float s_x[IN_FEATURES];
  const int tid = threadIdx.x;

  // Stage masked x into LDS: 256 threads x 4 float4 = 4096 floats.
  {
    const float4* src = (const float4*)xm;
    float4* dst = (float4*)s_x;
#pragma unroll
    for (int j = 0; j < 4; ++j) {
      int i = tid + j * 256;
      dst[i] = src[i];
    }
  }
  __syncthreads();

  const int lane = tid & 31;
  const int wave = tid >> 5;
  const int m0 = blockIdx.x * 128 + wave * 16;
  const int row = m0 + (lane & 15);
  const int koff = (lane >> 4) * 2;            // upper half-wave supplies K+2,K+3
  const float* wrow = W + (size_t)row * IN_FEATURES + koff;
  const float* xrow = s_x + koff;

  v8f c0 = {};
  v8f c1 = {};
#pragma unroll 4
  for (int k = 0; k < IN_FEATURES; k += 8) {
    v2f a0 = *(const v2f*)(wrow + k);
    v2f b0 = *(const v2f*)(xrow + k);
    c0 = __builtin_amdgcn_wmma_f32_16x16x4_f32(false, a0, false, b0,
                                               (short)0, c0, false, false);
    v2f a1 = *(const v2f*)(wrow + k + 4);
    v2f b1 = *(const v2f*)(xrow + k + 4);
    c1 = __builtin_amdgcn_wmma_f32_16x16x4_f32(false, a1, false, b1,
                                               (short)0, c1, false, false);
  }
  v8f c = c0 + c1;

  // C/D layout: VGPR r holds M = r (lanes 0-15) and M = r+8 (lanes 16-31);
  // all columns identical, so lanes 0 and 16 carry the 16 results.
  if ((lane & 15) == 0) {
    const int mbase = m0 + (lane >> 4) * 8;
    const float4 bb0 = *(const float4*)(bias + mbase);
    const float4 bb1 = *(const float4*)(bias + mbase + 4);
    float4 r0, r1;
    r0.x = c[0] + bb0.x; r0.y = c[1] + bb0.y;
    r0.z = c[2] + bb0.z; r0.w = c[3] + bb0.w;
    r1.x = c[4] + bb1.x; r1.y = c[5] + bb1.y;
    r1.z = c[6] + bb1.z; r1.w = c[7] + bb1.w;
    *(float4*)(out + mbase) = r0;
    *(float4*)(out + mbase + 4) = r1;
  }
}

extern "C" void kernel_launch(void* const* d_in, const int* in_sizes, int n_in,
                              void* d_out, int out_size, void* d_ws, size_t ws_size,
                              hipStream_t stream) {
  const float* x = (const float*)d_in[0];      // (1,1,4096) f32
  const float* W = (const float*)d_in[1];      // (14336,4096) f32
  const float* bias = (const float*)d_in[2];   // (14336,) f32
  float* out = (float*)d_out;                  // (14336,) f32
  float* xm = (float*)d_ws;                    // 4096 f32 masked-x scratch

  topk_mask_kernel<<<1, 1024, 0, stream>>>(x, xm);

  const int rows_per_block = 128;              // 8 waves * 16 rows
  gemv_wmma_kernel<<<OUT_FEATURES / rows_per_block, 256, 0, stream>>>(W, bias, xm, out);
}